// EdgeEmbedding_39539468927633
// MI455X (gfx1250) — compile-verified
//
#include <hip/hip_runtime.h>
#include <math.h>

typedef __attribute__((ext_vector_type(2))) float v2f;
typedef __attribute__((ext_vector_type(4))) float v4f;
typedef __attribute__((ext_vector_type(8))) float v8f;

#define NATOMS 2048
#define NBASIS 20
#define TILES_PER_ROW (NATOMS / 16)   // 128
#define WAVES_PER_BLOCK 8

// d_out layout (float32, reference return order, flattened):
//   [0, NN*20)        dist_edge [N][N][20]
//   [NN*20, NN*23)    dir_edge  [N][N][3]
//   [NN*23, NN*24)    mask      [N][N]   (1.0f / 0.0f)
__global__ __launch_bounds__(256) void edge_embed_kernel(const float* __restrict__ pos,
                                                         float* __restrict__ out)
{
    const int lane = threadIdx.x & 31;
    const int wave = threadIdx.x >> 5;
    const int tile = blockIdx.x * WAVES_PER_BLOCK + wave;   // 0 .. 16383
    const int ti   = tile >> 7;                             // tile / 128
    const int tj   = tile & (TILES_PER_ROW - 1);            // tile % 128

    const int  l15 = lane & 15;
    const bool hi  = lane >= 16;

    // ---- WMMA operands for D = A(16x4) x B(4x16), K=3 zero-padded to 4 ----
    // 32-bit A 16x4 layout: lanes 0-15 hold {K0,K1} in VGPR0/1, lanes 16-31 hold {K2,K3}.
    // B (4x16) mirrored: N striped across lanes, K across VGPRs / lane halves.
    const float* pA = pos + (size_t)(ti * 16 + l15) * 3;
    const float* pB = pos + (size_t)(tj * 16 + l15) * 3;
    const float ax = pA[0], ay = pA[1], az = pA[2];
    const float bx = pB[0], by = pB[1], bz = pB[2];

    v2f A, B;
    A.x = hi ? az : ax;  A.y = hi ? 0.0f : ay;
    B.x = hi ? bz : bx;  B.y = hi ? 0.0f : by;

    v8f C = {};
    // args: (neg_a, A, neg_b, B, c_mod, C, reuse_a, reuse_b)
    v8f D = __builtin_amdgcn_wmma_f32_16x16x4_f32(false, A, false, B, (short)0, C,
                                                  false, false);
    // D layout: VGPR r, lanes 0-15 -> (M=r, N=lane); lanes 16-31 -> (M=8+r, N=lane-16)

    // per-lane fixed column j; pB already points at pos[j]
    const int   j   = tj * 16 + l15;
    const float pjx = bx, pjy = by, pjz = bz;
    const float nj  = pjx * pjx + pjy * pjy + pjz * pjz;

    const size_t NN = (size_t)NATOMS * NATOMS;
    float* dist_edge = out;
    float* dir_edge  = out + NN * NBASIS;
    float* mask_out  = out + NN * (NBASIS + 3);

    const float kPi   = 3.14159265358979323846f;
    const float kInvR = 0.2f;    // 1 / CUTOFF

    #pragma unroll
    for (int r = 0; r < 8; ++r) {
        const int i = ti * 16 + (hi ? 8 + r : r);

        const float pix = pos[(size_t)i * 3 + 0];
        const float piy = pos[(size_t)i * 3 + 1];
        const float piz = pos[(size_t)i * 3 + 2];
        const float ni  = pix * pix + piy * piy + piz * piz;

        // squared distance from the WMMA Gram entry
        const float d2 = fmaxf(ni + nj - 2.0f * D[r], 0.0f);
        const bool  m  = (d2 < 25.0f) && (i != j);

        // one v_rsq_f32 gives everything: 1/dist, dist, 1/x
        const float rsq  = __builtin_amdgcn_rsqf(d2);     // 1/dist
        const float dist = d2 * rsq;                      // dist
        const float x    = m ? dist * kInvR : kInvR;      // safe_dist / 5 (safe=1 -> x=0.2)
        const float invx = m ? 5.0f * rsq : 5.0f;         // 1/x

        // PolynomialCutoff p=9: 1 - 55 x^9 + 99 x^10 - 45 x^11
        const float x2 = x * x;
        const float x4 = x2 * x2;
        const float x9 = x4 * x4 * x;
        const float env = fmaf(x9, fmaf(x, fmaf(x, -45.0f, 99.0f), -55.0f), 1.0f);

        // Bessel basis sin(k*pi*x)/x via Chebyshev recurrence: 1 sin + 1 cos + FMAs
        const float t    = kPi * x;
        const float s1   = __sinf(t);
        const float c1   = __cosf(t);
        const float twoc = 2.0f * c1;
        const float scale = m ? env * invx : 0.0f;

        float vals[NBASIS];
        float s_prev = 0.0f, s = s1;
        #pragma unroll
        for (int k = 0; k < NBASIS; ++k) {
            vals[k] = s * scale;
            const float s_next = fmaf(twoc, s, -s_prev);
            s_prev = s;
            s = s_next;
        }

        // dist_edge: 80 contiguous bytes per pair, 16B aligned -> 5x b128 stores
        float* dp = dist_edge + ((size_t)i * NATOMS + (size_t)j) * NBASIS;
        #pragma unroll
        for (int g = 0; g < 5; ++g) {
            v4f v;
            v.x = vals[4 * g + 0];
            v.y = vals[4 * g + 1];
            v.z = vals[4 * g + 2];
            v.w = vals[4 * g + 3];
            ((v4f*)dp)[g] = v;
        }

        // dir_edge: (pos[i]-pos[j]) / dist, masked; 1/dist is just rsq
        float* rp = dir_edge + ((size_t)i * NATOMS + (size_t)j) * 3;
        rp[0] = m ? (pix - pjx) * rsq : 0.0f;
        rp[1] = m ? (piy - pjy) * rsq : 0.0f;
        rp[2] = m ? (piz - pjz) * rsq : 0.0f;

        mask_out[(size_t)i * NATOMS + (size_t)j] = m ? 1.0f : 0.0f;
    }
}

extern "C" void kernel_launch(void* const* d_in, const int* in_sizes, int n_in,
                              void* d_out, int out_size, void* d_ws, size_t ws_size,
                              hipStream_t stream) {
    const float* pos = (const float*)d_in[0];      // [2048, 3] float32
    float* out = (float*)d_out;                    // 2048*2048*24 float32

    const int tiles  = TILES_PER_ROW * TILES_PER_ROW;      // 16384
    const int blocks = tiles / WAVES_PER_BLOCK;            // 2048
    edge_embed_kernel<<<blocks, 256, 0, stream>>>(pos, out);
}